// NearestNeighborRatio_83665962926606
// MI455X (gfx1250) — compile-verified
//
#include <hip/hip_runtime.h>

// NearestNeighborRatio on MI455X (gfx1250, wave32):
//   fused cdist^2 -> top-2 -> ratio test; cost matrix never touches memory.
//   GEMM via V_WMMA_F32_16X16X4_F32 (matches reference fp32 math exactly).
//   B (desc2) streams global -> LDS via GLOBAL_LOAD_ASYNC_TO_LDS_B128.
//   ||desc2||^2 precomputed once (prologue kernel) -> d_ws; inner loop keys
//   top-2 on t = ||b||^2 - 2ab (same ordering as d^2, row-constant dropped).
//   B LDS reads software-pipelined one K-step ahead of their WMMA consumers.

typedef float v2f __attribute__((ext_vector_type(2)));
typedef float v8f __attribute__((ext_vector_type(8)));
typedef int   v4i __attribute__((vector_size(16)));   // matches builtin param

#define ROWS_PER_BLOCK 64
#define NTILE          32      // columns staged per round (2 wave-groups x 16)
#define KDIM           128
#define LDSPITCH       132     // 128 + 4 pad floats: kills LDS bank conflicts

#define GLOBAL_AS __attribute__((address_space(1)))
#define LDS_AS    __attribute__((address_space(3)))

// ---- CDNA5 async global->LDS copy (no VGPR staging) ------------------------
#if __has_builtin(__builtin_amdgcn_global_load_async_to_lds_b128)
#define ASYNC_CP128(gp, lp)                                                    \
    __builtin_amdgcn_global_load_async_to_lds_b128(                            \
        (GLOBAL_AS v4i*)(gp), (LDS_AS v4i*)(lp), 0, 0)
#else
#define ASYNC_CP128(gp, lp)                                                    \
    asm volatile("global_load_async_to_lds_b128 %0, %1, off"                   \
                 :: "v"((LDS_AS v4i*)(lp)), "v"((GLOBAL_AS v4i*)(gp))          \
                 : "memory")
#endif

#if __has_builtin(__builtin_amdgcn_s_wait_asynccnt)
#define WAIT_ASYNC() __builtin_amdgcn_s_wait_asynccnt(0)
#else
#define WAIT_ASYNC() asm volatile("s_wait_asynccnt 0" ::: "memory")
#endif

// ---- prologue: ||desc2_row||^2, one thread per row -------------------------
__global__ __launch_bounds__(256)
void sqnorm_kernel(const float* __restrict__ d, float* __restrict__ out, int n)
{
    int i = blockIdx.x * 256 + threadIdx.x;
    if (i >= n) return;
    const float4* p = (const float4*)(d + (size_t)i * KDIM);
    float s = 0.f;
    #pragma unroll
    for (int k = 0; k < KDIM / 4; ++k) {
        float4 v = p[k];
        s += v.x * v.x + v.y * v.y + v.z * v.z + v.w * v.w;
    }
    out[i] = s;
}

__global__ __launch_bounds__(256)
void nn_ratio_wmma_kernel(const float* __restrict__ desc1,
                          const float* __restrict__ desc2,
                          const float* __restrict__ sqn2,   // ||desc2_j||^2
                          float* __restrict__ out,
                          int B1, int B2)
{
    __shared__ float sB[2][NTILE * LDSPITCH];          // double-buffered B tile
    __shared__ float sMerge[2][ROWS_PER_BLOCK][3];     // per-group top-2 staging

    const int tid   = threadIdx.x;
    const int w     = tid >> 5;      // wave 0..7
    const int L     = tid & 31;      // lane in wave32
    const int h     = L >> 4;        // half-wave (K-pair select in WMMA layout)
    const int ln    = L & 15;        // M index (A) / N index (B,C)
    const int g     = w >> 2;        // N-group: which 16 of the 32 staged cols
    const int strip = w & 3;         // 16-row strip inside the block

    const int rowBase = blockIdx.x * ROWS_PER_BLOCK;
    const int r       = rowBase + strip * 16 + ln;

    // per-thread slice of the cooperative 32x128-float tile copy:
    // 256 threads x 4 b128 = 1024 float4
    const int cp_rw[4] = { (tid + 0)   >> 5, (tid + 256) >> 5,
                           (tid + 512) >> 5, (tid + 768) >> 5 };
    const int cp_c4    = tid & 31;   // (f & 31) is invariant in f = tid + 256*jj

    // ---- A tile (16 rows x 128) resident in registers, WMMA A layout -------
    // lane L holds row (L&15); half h supplies K = 4s + 2h + {0,1}
    v2f   a[32];
    float sqa_p = 0.f;
    {
        const float* Ap = desc1 + (size_t)r * KDIM + 2 * h;
        #pragma unroll
        for (int s = 0; s < 32; ++s) {
            a[s] = *(const v2f*)(Ap + 4 * s);
            sqa_p += a[s].x * a[s].x + a[s].y * a[s].y;
        }
    }
    // full ||a_row||^2 then regather to C-matrix row mapping (row = v + 8h)
    float sqa_full = sqa_p + __shfl_xor(sqa_p, 16);
    float sqa[8];
    #pragma unroll
    for (int v = 0; v < 8; ++v)
        sqa[v] = __shfl(sqa_full, v + 8 * h);

    // ---- running top-2 on key t = ||b||^2 - 2ab (row-const ||a||^2 dropped;
    //      identical ordering to d^2) -----------------------------------------
    float b1d[8], b2d[8];
    int   b1i[8];
    #pragma unroll
    for (int v = 0; v < 8; ++v) { b1d[v] = 3.4e38f; b2d[v] = 3.4e38f; b1i[v] = 0; }

    const int nrounds = B2 / NTILE;

    // ---- prime buffer 0 via async copy -------------------------------------
    #pragma unroll
    for (int jj = 0; jj < 4; ++jj) {
        const float* gp = desc2 + (size_t)cp_rw[jj] * KDIM + cp_c4 * 4;
        float*       lp = &sB[0][cp_rw[jj] * LDSPITCH + cp_c4 * 4];
        ASYNC_CP128(gp, lp);
    }
    WAIT_ASYNC();
    __syncthreads();

    for (int j = 0; j < nrounds; ++j) {
        const int cur = j & 1;

        if (j + 1 < nrounds) {   // async-prefetch next 32 columns into buf^1
            const int nxt = cur ^ 1;
            #pragma unroll
            for (int jj = 0; jj < 4; ++jj) {
                const float* gp = desc2 +
                    ((size_t)(j + 1) * NTILE + cp_rw[jj]) * KDIM + cp_c4 * 4;
                float* lp = &sB[nxt][cp_rw[jj] * LDSPITCH + cp_c4 * 4];
                ASYNC_CP128(gp, lp);
            }
        }

        const int   coln = j * NTILE + g * 16 + ln;
        const float sqb  = sqn2[coln];        // precomputed, L2-resident

        // ---- 16x16 tile: 32 V_WMMA_F32_16X16X4_F32 over K=128 --------------
        // two accumulators halve the WMMA->WMMA dependency chain; B LDS reads
        // issued one K-step ahead of their consumers to hide ds latency.
        v8f acc0 = {0.f,0.f,0.f,0.f,0.f,0.f,0.f,0.f};
        v8f acc1 = {0.f,0.f,0.f,0.f,0.f,0.f,0.f,0.f};
        const float* Bp = &sB[cur][(g * 16 + ln) * LDSPITCH + 2 * h];

        v2f c0 = *(const v2f*)(Bp + 0);
        v2f c1 = *(const v2f*)(Bp + 4);
        #pragma unroll
        for (int s = 0; s < 30; s += 2) {
            v2f n0 = *(const v2f*)(Bp + 4 * s + 8);
            v2f n1 = *(const v2f*)(Bp + 4 * s + 12);
            acc0 = __builtin_amdgcn_wmma_f32_16x16x4_f32(
                       false, a[s],     false, c0, (short)0, acc0, false, false);
            acc1 = __builtin_amdgcn_wmma_f32_16x16x4_f32(
                       false, a[s + 1], false, c1, (short)0, acc1, false, false);
            c0 = n0; c1 = n1;
        }
        acc0 = __builtin_amdgcn_wmma_f32_16x16x4_f32(
                   false, a[30], false, c0, (short)0, acc0, false, false);
        acc1 = __builtin_amdgcn_wmma_f32_16x16x4_f32(
                   false, a[31], false, c1, (short)0, acc1, false, false);

        // branchless top-2 update: 2 cmp + 4 cndmask per row, no min/canon ops
        #pragma unroll
        for (int v = 0; v < 8; ++v) {
            float dot = acc0[v] + acc1[v];
            float t   = __builtin_fmaf(-2.f, dot, sqb);
            bool  c1b = t < b1d[v];
            bool  c2b = t < b2d[v];
            float nb2 = c1b ? b1d[v] : (c2b ? t : b2d[v]);
            b1d[v] = c1b ? t    : b1d[v];
            b1i[v] = c1b ? coln : b1i[v];
            b2d[v] = nb2;
        }

        WAIT_ASYNC();            // drain async copies before publishing buf^1
        __syncthreads();
    }

    // ---- top-2 merge across the 16 lanes sharing each output row -----------
    #pragma unroll
    for (int m = 1; m < 16; m <<= 1) {
        #pragma unroll
        for (int v = 0; v < 8; ++v) {
            float ob1 = __shfl_xor(b1d[v], m);
            int   oi1 = __shfl_xor(b1i[v], m);
            float ob2 = __shfl_xor(b2d[v], m);
            if (ob1 < b1d[v]) { b2d[v] = fminf(b1d[v], ob2); b1d[v] = ob1; b1i[v] = oi1; }
            else              { b2d[v] = fminf(b2d[v], ob1); }
        }
    }

    // ---- stage per-wave results; re-add ||a||^2 and clamp to d^2 >= 0 ------
    if (ln == 0) {
        #pragma unroll
        for (int v = 0; v < 8; ++v) {
            int lr = strip * 16 + 8 * h + v;
            sMerge[g][lr][0] = fmaxf(sqa[v] + b1d[v], 0.f);
            sMerge[g][lr][1] = __int_as_float(b1i[v]);
            sMerge[g][lr][2] = fmaxf(sqa[v] + b2d[v], 0.f);
        }
    }
    __syncthreads();

    // ---- merge the two N-groups, ratio test, write packed outputs ----------
    if (tid < ROWS_PER_BLOCK) {
        float x1 = sMerge[0][tid][0], x2 = sMerge[0][tid][2];
        int   xi = __float_as_int(sMerge[0][tid][1]);
        float y1 = sMerge[1][tid][0], y2 = sMerge[1][tid][2];
        int   yi = __float_as_int(sMerge[1][tid][1]);
        float f1, f2; int fi;
        if (y1 < x1) { f1 = y1; fi = yi; f2 = fminf(x1, y2); }
        else         { f1 = x1; fi = xi; f2 = fminf(x2, y1); }

        float ratio = sqrtf(f1 / f2);          // sqrt(d1^2/d2^2) = d1/d2
        bool  mask  = ratio <= 0.8f;
        int   R     = rowBase + tid;

        out[R]              = mask ? ratio : 0.f;   // match_dists
        out[B1 + 2 * R + 0] = (float)R;             // matches_idxs[:,0]
        out[B1 + 2 * R + 1] = (float)fi;            // matches_idxs[:,1]
        out[B1 * 3 + R]     = mask ? 1.f : 0.f;     // mask
    }
}

extern "C" void kernel_launch(void* const* d_in, const int* in_sizes, int n_in,
                              void* d_out, int out_size, void* d_ws, size_t ws_size,
                              hipStream_t stream) {
    const float* desc1 = (const float*)d_in[0];
    const float* desc2 = (const float*)d_in[1];
    float*       out   = (float*)d_out;
    float*       sqn2  = (float*)d_ws;   // 16384 floats of scratch
    const int B1 = in_sizes[0] / KDIM;   // 16384
    const int B2 = in_sizes[1] / KDIM;   // 16384

    hipLaunchKernelGGL(sqnorm_kernel, dim3((B2 + 255) / 256), dim3(256),
                       0, stream, desc2, sqn2, B2);
    hipLaunchKernelGGL(nn_ratio_wmma_kernel, dim3(B1 / ROWS_PER_BLOCK), dim3(256),
                       0, stream, desc1, desc2, sqn2, out, B1, B2);
}